// ModulatedConv2d_31293131719001
// MI455X (gfx1250) — compile-verified
//
#include <hip/hip_runtime.h>
#include <math.h>

#define IN_C     512
#define OUT_C    512
#define KSZ      3
#define STYLE_D  512
#define BATCH    16
#define HH       64
#define WW       64

#define OBLK     64             // out channels per workgroup
#define ICSTEP   16             // in channels per K-chunk
#define ASTRIDE  20             // padded i-stride per (tap,o) in weight LDS
#define BSTRIDE  18             // padded ch-stride per (row,col) in input LDS
#define NROWS    4              // input rows staged: h-1 .. h+2

typedef __attribute__((ext_vector_type(2))) float v2f;
typedef __attribute__((ext_vector_type(8))) float v8f;

typedef __attribute__((address_space(1))) int gint_t;   // global int
typedef __attribute__((address_space(3))) int lint_t;   // LDS int

#if __has_builtin(__builtin_amdgcn_global_load_async_to_lds_b32)
#define HAVE_ASYNC_LDS 1
#else
#define HAVE_ASYNC_LDS 0
#endif

__device__ __forceinline__ void wait_async0() {
#if __has_builtin(__builtin_amdgcn_s_wait_asynccnt)
  __builtin_amdgcn_s_wait_asynccnt(0);
#else
  asm volatile("s_wait_asynccnt 0x0" ::: "memory");
#endif
}

// ---------------------------------------------------------------------------
// Kernel 1: style modulation  s[b][i] = lin_scale * (style[b] . mod_w[i]) + mod_b[i]
// ---------------------------------------------------------------------------
__global__ __launch_bounds__(256) void modstyle_kernel(
    const float* __restrict__ style, const float* __restrict__ mod_w,
    const float* __restrict__ mod_b, float* __restrict__ s_out) {
  const int tid = blockIdx.x * 256 + threadIdx.x;      // 8192 = 16*512
  const int b = tid >> 9;
  const int i = tid & 511;
  const float lin_scale = 0.04419417382415922f;        // 1/sqrt(512)
  const float* st = style + (size_t)b * STYLE_D;
  const float* mw = mod_w + (size_t)i * STYLE_D;
  float acc = 0.f;
  for (int j = 0; j < STYLE_D; j += 4) {
    acc += st[j] * mw[j] + st[j + 1] * mw[j + 1] +
           st[j + 2] * mw[j + 2] + st[j + 3] * mw[j + 3];
  }
  s_out[tid] = acc * lin_scale + mod_b[i];
}

// ---------------------------------------------------------------------------
// Kernel 2: demodulation scale  D[b][o] = cs * rsqrt(cs^2 * sum_i s^2 * sum_t w^2 + 1e-8)
// ---------------------------------------------------------------------------
__global__ __launch_bounds__(256) void demod_kernel(
    const float* __restrict__ weight, const float* __restrict__ s,
    float* __restrict__ d_out) {
  const int tid = blockIdx.x * 256 + threadIdx.x;      // 8192 = 16*512
  const int b = tid >> 9;
  const int o = tid & 511;
  const float cs = 0.014731391274719739f;              // 1/sqrt(512*9)
  const float* wo = weight + (size_t)o * IN_C * 9;
  const float* sb = s + (size_t)b * IN_C;
  float acc = 0.f;
  for (int i = 0; i < IN_C; ++i) {
    const float sv = sb[i];
    const float* wp = wo + i * 9;
    float ws = 0.f;
#pragma unroll
    for (int t = 0; t < 9; ++t) { float v = wp[t]; ws += v * v; }
    acc += sv * sv * ws;
  }
  d_out[tid] = cs * rsqrtf(cs * cs * acc + 1e-8f);
}

// ---------------------------------------------------------------------------
// Kernel 3: implicit-GEMM conv with V_WMMA_F32_16X16X4_F32.
// Grid: (OUT_C/OBLK, HH/2, BATCH). Block = 256 threads = 8 waves.
// Workgroup tile: 64 out-ch x (2 rows x 64 px). Wave owns 32 out x 32 px.
// Style modulation is folded into the WEIGHT staging (b fixed per WG), so
// the input tile is a byte-exact copy -> staged with the CDNA5 async-to-LDS
// path (GLOBAL_LOAD_ASYNC_TO_LDS_B32, ASYNCcnt), EXEC-masked for the halo.
// ---------------------------------------------------------------------------
__global__ __launch_bounds__(256) void modconv_kernel(
    const float* __restrict__ x, const float* __restrict__ weight,
    const float* __restrict__ s, const float* __restrict__ dmod,
    float* __restrict__ out) {
  __shared__ float sw[9 * OBLK * ASTRIDE];          // 11520 floats: [tap][o][i pad 20]
  __shared__ float sx[NROWS * 66 * BSTRIDE];        //  4752 floats: [row][col][ch pad 18]

  const int o0   = blockIdx.x * OBLK;
  const int h0   = blockIdx.y * 2;                  // two output rows per WG
  const int b    = blockIdx.z;
  const int t    = threadIdx.x;
  const int lane = t & 31;
  const int wave = t >> 5;
  const int wo   = wave >> 2;                       // 0..1 : 32-out sub-block
  const int wp   = wave & 3;                        // 0..3 : 32-px sub-block
  const int prow = wp >> 1;                         // 0..1 : output row within tile
  const int pcol0 = (wp & 1) * 32;                  // 0 / 32 : column base

  const int m  = lane & 15;                         // M (A) / N (B,C) index
  const int kk = (lane >> 4) << 1;                  // 0 or 2 : K sub-index (ISA layout)

  v8f c00 = {}, c01 = {}, c10 = {}, c11 = {};       // [co][cp]

  const float* sb = s + (size_t)b * IN_C;

  // Pre-zero input buffer once: halo slots are chunk-invariant and are never
  // overwritten by the (EXEC-masked) async copies below.
  for (int f = t; f < NROWS * 66 * BSTRIDE; f += 256) sx[f] = 0.f;
  __syncthreads();

  for (int ic0 = 0; ic0 < IN_C; ic0 += ICSTEP) {
    // --- input tile: pure copy, issue async global->LDS first (long latency) ---
    for (int f = t; f < ICSTEP * NROWS * 66; f += 256) {
      const int ch  = f / (NROWS * 66);
      const int rem = f - ch * (NROWS * 66);
      const int row = rem / 66;
      const int col = rem - row * 66;
      const int hh  = h0 + row - 1;
      const int ww  = col - 1;
      if ((unsigned)hh < 64u && (unsigned)ww < 64u) {
        const float* gp = &x[((size_t)(b * IN_C + ic0 + ch) * HH + hh) * WW + ww];
        float* lp = &sx[(row * 66 + col) * BSTRIDE + ch];
#if HAVE_ASYNC_LDS
        __builtin_amdgcn_global_load_async_to_lds_b32(
            (gint_t*)gp, (lint_t*)lp, 0, 0);
#else
        *lp = *gp;
#endif
      }
    }
    // --- stage modulated weights: A = w * s[b][i]; 9216 elems, 36/thread ---
#pragma unroll
    for (int k = 0; k < 36; ++k) {
      const int f   = t + (k << 8);
      const int o_l = f / 144;                      // 144 = ICSTEP*9
      const int rem = f - o_l * 144;
      const int i_l = rem / 9;
      const int tap = rem - i_l * 9;
      sw[(tap * OBLK + o_l) * ASTRIDE + i_l] =
          weight[(size_t)(o0 + o_l) * (IN_C * 9) + (size_t)(ic0 + i_l) * 9 + tap] *
          sb[ic0 + i_l];
    }
    wait_async0();                                  // our async copies landed in LDS
    __syncthreads();                                // all waves' staging visible

#pragma unroll
    for (int kh = 0; kh < 3; ++kh) {
#pragma unroll
      for (int kw = 0; kw < 3; ++kw) {
        const int tap = kh * 3 + kw;
#pragma unroll
        for (int ks = 0; ks < 4; ++ks) {
          const int ib = ks * 4 + kk;               // K pair base for this lane half
          const v2f a0 = *reinterpret_cast<const v2f*>(
              &sw[(tap * OBLK + wo * 32 + m) * ASTRIDE + ib]);
          const v2f a1 = *reinterpret_cast<const v2f*>(
              &sw[(tap * OBLK + wo * 32 + 16 + m) * ASTRIDE + ib]);
          const int bbase = ((prow + kh) * 66 + pcol0 + m + kw) * BSTRIDE + ib;
          const v2f b0 = *reinterpret_cast<const v2f*>(&sx[bbase]);
          const v2f b1 = *reinterpret_cast<const v2f*>(&sx[bbase + 16 * BSTRIDE]);
          c00 = __builtin_amdgcn_wmma_f32_16x16x4_f32(false, a0, false, b0, (short)0, c00, false, false);
          c01 = __builtin_amdgcn_wmma_f32_16x16x4_f32(false, a0, false, b1, (short)0, c01, false, false);
          c10 = __builtin_amdgcn_wmma_f32_16x16x4_f32(false, a1, false, b0, (short)0, c10, false, false);
          c11 = __builtin_amdgcn_wmma_f32_16x16x4_f32(false, a1, false, b1, (short)0, c11, false, false);
        }
      }
    }
    __syncthreads();                                // done reading before next stage
  }

  // --- epilogue: apply demod scale, coalesced stores ---
  const int n    = lane & 15;
  const int hi8  = (lane >> 4) << 3;                // +8 rows for upper lane half
  const int hrow = h0 + prow;
#pragma unroll
  for (int co = 0; co < 2; ++co) {
#pragma unroll
    for (int r = 0; r < 8; ++r) {
      const int o = o0 + wo * 32 + co * 16 + hi8 + r;
      const float dm = dmod[b * OUT_C + o];
      const size_t base = ((size_t)(b * OUT_C + o) * HH + hrow) * WW + pcol0 + n;
      out[base]      = (co ? c10[r] : c00[r]) * dm;
      out[base + 16] = (co ? c11[r] : c01[r]) * dm;
    }
  }
}

// ---------------------------------------------------------------------------
extern "C" void kernel_launch(void* const* d_in, const int* in_sizes, int n_in,
                              void* d_out, int out_size, void* d_ws, size_t ws_size,
                              hipStream_t stream) {
  const float* x      = (const float*)d_in[0];
  const float* style  = (const float*)d_in[1];
  const float* weight = (const float*)d_in[2];
  const float* mod_w  = (const float*)d_in[3];
  const float* mod_b  = (const float*)d_in[4];
  float* out = (float*)d_out;

  float* s_ws  = (float*)d_ws;                 // 16*512 floats
  float* d_mod = s_ws + BATCH * IN_C;          // 16*512 floats

  modstyle_kernel<<<BATCH * IN_C / 256, 256, 0, stream>>>(style, mod_w, mod_b, s_ws);
  demod_kernel<<<BATCH * OUT_C / 256, 256, 0, stream>>>(weight, s_ws, d_mod);

  dim3 grid(OUT_C / OBLK, HH / 2, BATCH);      // (8, 32, 16)
  modconv_kernel<<<grid, 256, 0, stream>>>(x, weight, s_ws, d_mod, out);
}